// DenseGNN_41283225649173
// MI455X (gfx1250) — compile-verified
//
#include <hip/hip_runtime.h>
#include <hip/hip_bf16.h>
#include <math.h>

typedef __attribute__((ext_vector_type(2))) float v2f;
typedef __attribute__((ext_vector_type(8))) float v8f;

#define N_NODES 20000
#define N_EDGES 320000
#define IN_CH   128
#define HID     64
#define E_DIM   32
#define N_LAYERS 4
#define XLD     (HID * (N_LAYERS + 1))   // 320: dense-concat feature matrix stride

// ---------------------------------------------------------------------------
// Generic fp32 WMMA GEMM: D[rows,Fo] = act(A[rows,K] @ B[K,Fo] + bias) (+resid)
// One wave computes a 16x64 output tile (4 accumulators) via
// V_WMMA_F32_16X16X4_F32: one A-fragment load feeds 4 WMMAs.
// rows multiple of 16; Fo multiple of 64; K multiple of 4 (true in this model).
// act: 0=none 1=relu 2=sigmoid 3=gelu(exact)
// ---------------------------------------------------------------------------
__global__ __launch_bounds__(256) void wgemm_f32(
    const float* __restrict__ A, int lda,
    const float* __restrict__ B,            // [K, Fo] row-major
    const float* __restrict__ bias,         // [Fo] or null
    const float* __restrict__ resid, int ldr,  // same shape as D, or null
    float* __restrict__ D, int ldd,
    int rows, int K, int Fo, int act)
{
  const int wavesPerBlock = blockDim.x >> 5;
  const int wave = blockIdx.x * wavesPerBlock + (threadIdx.x >> 5);
  const int lane = threadIdx.x & 31;
  const int colGroups = Fo >> 6;            // 16x64 tile per wave
  const int rowTiles = rows >> 4;
  if (wave >= rowTiles * colGroups) return; // wave-uniform: EXEC stays all-ones
  const int rt = wave / colGroups;
  const int cg = wave - rt * colGroups;

  const int hf = lane >> 4;     // which K-pair this lane holds (K+0/1 vs K+2/3)
  const int r  = lane & 15;     // matrix row (A) / col (B,D) within tile

  const float* Arow = A + (size_t)(rt * 16 + r) * lda;
  const float* Bcol = B + (cg * 64 + r);

  v8f acc[4] = {{}, {}, {}, {}};
  for (int k = 0; k < K; k += 4) {
    const int ka = k + 2 * hf;
    const v2f a = *(const v2f*)(Arow + ka);        // A[r, ka], A[r, ka+1]
    const float* b0 = Bcol + (size_t)ka * Fo;
    const float* b1 = Bcol + (size_t)(ka + 1) * Fo;
#pragma unroll
    for (int j = 0; j < 4; ++j) {
      v2f b;
      b.x = b0[j * 16];                            // B[ka,   cg*64 + j*16 + r]
      b.y = b1[j * 16];                            // B[ka+1, cg*64 + j*16 + r]
      acc[j] = __builtin_amdgcn_wmma_f32_16x16x4_f32(
          false, a, false, b, (short)0, acc[j], false, false);
    }
  }

#pragma unroll
  for (int j = 0; j < 4; ++j) {
    const int col = cg * 64 + j * 16 + r;
    const float bv = bias ? bias[col] : 0.0f;
#pragma unroll
    for (int v = 0; v < 8; ++v) {
      const int row = rt * 16 + v + 8 * hf;        // C/D layout: VGPR v -> M=v / v+8
      float x = acc[j][v] + bv;
      if (act == 1)      x = fmaxf(x, 0.0f);
      else if (act == 2) x = 1.0f / (1.0f + expf(-x));
      else if (act == 3) x = 0.5f * x * (1.0f + erff(x * 0.70710678118654752f));
      if (resid) x += resid[(size_t)row * ldr + col];
      D[(size_t)row * ldd + col] = x;
    }
  }
}

// ---------------------------------------------------------------------------
// LayerNorm over rows, one wave per row. width in {64,128,192,256}.
// act: 0=none 1=relu 2=gelu(exact)
// ---------------------------------------------------------------------------
__global__ __launch_bounds__(256) void ln_rows(
    const float* __restrict__ in, int ldi,
    float* __restrict__ out, int ldo,
    const float* __restrict__ g, const float* __restrict__ b,
    int n, int width, int act)
{
  const int row = blockIdx.x * (blockDim.x >> 5) + (threadIdx.x >> 5);
  if (row >= n) return;
  const int lane = threadIdx.x & 31;
  const int per = width >> 5;                    // <= 8
  const float* x = in + (size_t)row * ldi;
  float vals[8];
  float s = 0.0f;
  for (int i = 0; i < per; ++i) { vals[i] = x[lane + 32 * i]; s += vals[i]; }
  for (int off = 16; off; off >>= 1) s += __shfl_xor(s, off, 32);
  const float mu = s / (float)width;
  float vs = 0.0f;
  for (int i = 0; i < per; ++i) { float d = vals[i] - mu; vs += d * d; }
  for (int off = 16; off; off >>= 1) vs += __shfl_xor(vs, off, 32);
  const float inv = rsqrtf(vs / (float)width + 1e-5f);
  float* o = out + (size_t)row * ldo;
  for (int i = 0; i < per; ++i) {
    const int c = lane + 32 * i;
    float y = (vals[i] - mu) * inv * g[c] + b[c];
    if (act == 1)      y = fmaxf(y, 0.0f);
    else if (act == 2) y = 0.5f * y * (1.0f + erff(y * 0.70710678118654752f));
    o[c] = y;
  }
}

// ---------------------------------------------------------------------------
// Edge pass 1: msg = relu(h_src[src] + edge_proj) + 1e-7 (written in place over
// edge_proj), and per-(dst,channel) running max via uint atomicMax (msg > 0).
// ---------------------------------------------------------------------------
__global__ __launch_bounds__(256) void edge_pass1(
    const float* __restrict__ HSRC, float* __restrict__ EP,
    const int* __restrict__ src, const int* __restrict__ dst,
    unsigned int* __restrict__ Mx, int total)
{
  const int tid = blockIdx.x * blockDim.x + threadIdx.x;
  if (tid >= total) return;
  const int e = tid >> 6, c = tid & 63;
  float m = HSRC[(size_t)src[e] * HID + c] + EP[tid];
  m = fmaxf(m, 0.0f) + 1e-7f;
  EP[tid] = m;
  atomicMax(&Mx[(size_t)dst[e] * HID + c], __float_as_uint(m));
}

// Edge pass 2: e = exp(msg - max); S += e; T += e*msg (per dst,channel).
__global__ __launch_bounds__(256) void edge_pass2(
    const float* __restrict__ EP, const int* __restrict__ dst,
    const float* __restrict__ Mx, float* __restrict__ S, float* __restrict__ T,
    int total)
{
  const int tid = blockIdx.x * blockDim.x + threadIdx.x;
  if (tid >= total) return;
  const int e = tid >> 6, c = tid & 63;
  const size_t di = (size_t)dst[e] * HID + c;
  const float msg = EP[tid];
  const float ex = expf(msg - Mx[di]);
  atomicAdd(&S[di], ex);
  atomicAdd(&T[di], ex * msg);
}

// MsgNorm + residual: agg = T/(S+1e-16); agg /= ||agg||; out = agg*||hdst||*sc + hdst
__global__ __launch_bounds__(256) void msgnorm_k(
    const float* __restrict__ T, const float* __restrict__ S,
    const float* __restrict__ HDST, int ldh,
    const float* __restrict__ msg_scale,
    float* __restrict__ OUT, int n)
{
  const int row = blockIdx.x * (blockDim.x >> 5) + (threadIdx.x >> 5);
  if (row >= n) return;
  const int lane = threadIdx.x & 31;
  const size_t base = (size_t)row * HID;
  const size_t hb = (size_t)row * ldh;
  const float a0 = T[base + lane]      / (S[base + lane]      + 1e-16f);
  const float a1 = T[base + lane + 32] / (S[base + lane + 32] + 1e-16f);
  const float h0 = HDST[hb + lane], h1 = HDST[hb + lane + 32];
  float ss = a0 * a0 + a1 * a1;
  float hh = h0 * h0 + h1 * h1;
  for (int off = 16; off; off >>= 1) {
    ss += __shfl_xor(ss, off, 32);
    hh += __shfl_xor(hh, off, 32);
  }
  const float sc = rsqrtf(ss + 1e-24f) * sqrtf(hh) * msg_scale[0];
  OUT[base + lane]      = a0 * sc + h0;
  OUT[base + lane + 32] = a1 * sc + h1;
}

// h = g*conv + (1-g)*skip
__global__ __launch_bounds__(256) void gate_combine(
    const float* __restrict__ G, const float* __restrict__ CONV,
    const float* __restrict__ SK, int ldsk, float* __restrict__ HB, int total)
{
  const int tid = blockIdx.x * blockDim.x + threadIdx.x;
  if (tid >= total) return;
  const int row = tid >> 6, c = tid & 63;
  const float g = G[tid];
  HB[tid] = g * CONV[tid] + (1.0f - g) * SK[(size_t)row * ldsk + c];
}

// gcat = [conv_out | residual] into [N,128]
__global__ __launch_bounds__(256) void pack_gcat(
    const float* __restrict__ CONV, const float* __restrict__ X, int ldx, int xoff,
    float* __restrict__ G2, int total)
{
  const int tid = blockIdx.x * blockDim.x + threadIdx.x;
  if (tid >= total) return;
  const int row = tid >> 7, c = tid & 127;
  G2[tid] = (c < HID) ? CONV[(size_t)row * HID + c]
                      : X[(size_t)row * ldx + xoff + (c - HID)];
}

__global__ __launch_bounds__(256) void fill_zero(float* __restrict__ p, int n) {
  const int i = blockIdx.x * blockDim.x + threadIdx.x;
  if (i < n) p[i] = 0.0f;
}

__global__ __launch_bounds__(256) void copy_cols(
    const float* __restrict__ X, int ldx, int off, float* __restrict__ out, int total)
{
  const int tid = blockIdx.x * blockDim.x + threadIdx.x;
  if (tid >= total) return;
  const int r = tid >> 6, c = tid & 63;
  out[tid] = X[(size_t)r * ldx + off + c];
}

// ---------------------------------------------------------------------------
// Host orchestration
// ---------------------------------------------------------------------------
struct LayerP {
  const float *n1_g, *n1_b, *lin_edge, *msg_scale;
  const float *mlp_w1, *mlp_ln_g, *mlp_ln_b, *mlp_w2;
  const float *gate_w1, *gate_b1, *gate_ln_g, *gate_ln_b, *gate_w2, *gate_b2;
  const float *n2_g, *n2_b, *ffn_w1, *ffn_b1, *ffn_w2, *ffn_b2;
  const float *lin_src, *lin_dst, *skip_w, *skip_b;
};

static inline void gemm(const float* A, int lda, const float* B, const float* bias,
                        const float* resid, int ldr, float* D, int ldd,
                        int rows, int K, int Fo, int act, hipStream_t s)
{
  const int tiles = (rows >> 4) * (Fo >> 6);   // one 16x64 tile per wave
  wgemm_f32<<<(tiles + 7) / 8, 256, 0, s>>>(A, lda, B, bias, resid, ldr, D, ldd,
                                            rows, K, Fo, act);
}

extern "C" void kernel_launch(void* const* d_in, const int* in_sizes, int n_in,
                              void* d_out, int out_size, void* d_ws, size_t ws_size,
                              hipStream_t stream) {
  (void)in_sizes; (void)n_in; (void)out_size; (void)ws_size;
  int idx = 0;
  auto nf = [&]() { return (const float*)d_in[idx++]; };

  const float* x          = nf();
  const int*   edge_index = (const int*)d_in[idx++];
  const float* edge_attr  = nf();
  const float* in_w       = nf();
  const float* in_b       = nf();

  LayerP L[N_LAYERS];
  for (int i = 0; i < N_LAYERS; ++i) {
    LayerP& p = L[i];
    p.n1_g = nf(); p.n1_b = nf(); p.lin_edge = nf(); p.msg_scale = nf();
    p.mlp_w1 = nf(); p.mlp_ln_g = nf(); p.mlp_ln_b = nf(); p.mlp_w2 = nf();
    p.gate_w1 = nf(); p.gate_b1 = nf(); p.gate_ln_g = nf(); p.gate_ln_b = nf();
    p.gate_w2 = nf(); p.gate_b2 = nf(); p.n2_g = nf(); p.n2_b = nf();
    p.ffn_w1 = nf(); p.ffn_b1 = nf(); p.ffn_w2 = nf(); p.ffn_b2 = nf();
    if (i > 0) { p.lin_src = nf(); p.lin_dst = nf(); p.skip_w = nf(); p.skip_b = nf(); }
    else       { p.lin_src = p.lin_dst = p.skip_w = p.skip_b = nullptr; }
  }

  const int* src = edge_index;            // row 0 of [2,E]
  const int* dst = edge_index + N_EDGES;  // row 1

  // Workspace layout (floats)
  float* W = (float*)d_ws;
  size_t off = 0;
  auto alloc = [&](size_t n) { float* p = W + off; off += n; return p; };
  float* X      = alloc((size_t)N_NODES * XLD);        // dense-concat features
  float* CUR    = alloc((size_t)N_NODES * 256);        // layernormed concat
  float* HSRC   = alloc((size_t)N_NODES * HID);
  float* HDST   = alloc((size_t)N_NODES * HID);
  float* EP     = alloc((size_t)N_EDGES * HID);        // edge proj -> msg (in place)
  float* Mx     = alloc((size_t)N_NODES * HID);        // softmax max
  float* Sm     = alloc((size_t)N_NODES * HID);        // softmax denom
  float* Tm     = alloc((size_t)N_NODES * HID);        // weighted numerator
  float* OUTb   = alloc((size_t)N_NODES * HID);
  float* TMP128 = alloc((size_t)N_NODES * 128);
  float* CONV   = alloc((size_t)N_NODES * HID);
  float* SKIP   = alloc((size_t)N_NODES * HID);
  float* Gg     = alloc((size_t)N_NODES * HID);
  float* G2     = alloc((size_t)N_NODES * HID);
  float* HB     = alloc((size_t)N_NODES * HID);
  float* TMP256 = alloc((size_t)N_NODES * 256);

  const int nodeElems = N_NODES * HID;                 // 1,280,000
  const int edgeElems = N_EDGES * HID;                 // 20,480,000
  const int rowWaveBlocks = (N_NODES + 7) / 8;         // ln/msgnorm: 8 rows/block

  // h0 = x @ in_w + in_b  -> X[:, 0:64]
  gemm(x, IN_CH, in_w, in_b, nullptr, 0, X, XLD, N_NODES, IN_CH, HID, 0, stream);

  for (int i = 0; i < N_LAYERS; ++i) {
    const LayerP& p = L[i];
    const int ci = HID * (i + 1);

    // cur = LayerNorm(concat(xs))  (X[:, :ci] -> CUR[N, ci])
    ln_rows<<<rowWaveBlocks, 256, 0, stream>>>(X, XLD, CUR, ci, p.n1_g, p.n1_b,
                                               N_NODES, ci, 0);

    const float* hsrc = CUR; const float* hdst = CUR; int ldh = ci;
    if (i > 0) {
      gemm(CUR, ci, p.lin_src, nullptr, nullptr, 0, HSRC, HID, N_NODES, ci, HID, 0, stream);
      gemm(CUR, ci, p.lin_dst, nullptr, nullptr, 0, HDST, HID, N_NODES, ci, HID, 0, stream);
      hsrc = HSRC; hdst = HDST; ldh = HID;
    }

    // edge projection: EP = edge_attr @ lin_edge
    gemm(edge_attr, E_DIM, p.lin_edge, nullptr, nullptr, 0, EP, HID,
         N_EDGES, E_DIM, HID, 0, stream);

    // scatter softmax aggregation (2-pass)
    fill_zero<<<(3 * nodeElems + 255) / 256, 256, 0, stream>>>(Mx, 3 * nodeElems);
    edge_pass1<<<(edgeElems + 255) / 256, 256, 0, stream>>>(
        hsrc, EP, src, dst, (unsigned int*)Mx, edgeElems);
    edge_pass2<<<(edgeElems + 255) / 256, 256, 0, stream>>>(
        EP, dst, Mx, Sm, Tm, edgeElems);
    msgnorm_k<<<rowWaveBlocks, 256, 0, stream>>>(Tm, Sm, hdst, ldh, p.msg_scale,
                                                 OUTb, N_NODES);

    // conv MLP: Linear -> LN -> ReLU -> Linear
    gemm(OUTb, HID, p.mlp_w1, nullptr, nullptr, 0, TMP128, 128, N_NODES, HID, 128, 0, stream);
    ln_rows<<<rowWaveBlocks, 256, 0, stream>>>(TMP128, 128, TMP128, 128,
                                               p.mlp_ln_g, p.mlp_ln_b, N_NODES, 128, 1);
    gemm(TMP128, 128, p.mlp_w2, nullptr, nullptr, 0, CONV, HID, N_NODES, 128, HID, 0, stream);

    // skip path
    const float* skipp = CUR; int ldsk = ci;
    if (i > 0) {
      gemm(CUR, ci, p.skip_w, p.skip_b, nullptr, 0, SKIP, HID, N_NODES, ci, HID, 0, stream);
      skipp = SKIP; ldsk = HID;
    }

    // gate: sigmoid(gelu(LN([conv|residual] @ W1 + b1)) @ W2 + b2)
    pack_gcat<<<(N_NODES * 128 + 255) / 256, 256, 0, stream>>>(
        CONV, X, XLD, HID * i, TMP128, N_NODES * 128);
    gemm(TMP128, 128, p.gate_w1, p.gate_b1, nullptr, 0, Gg, HID, N_NODES, 128, HID, 0, stream);
    ln_rows<<<rowWaveBlocks, 256, 0, stream>>>(Gg, HID, Gg, HID,
                                               p.gate_ln_g, p.gate_ln_b, N_NODES, HID, 2);
    gemm(Gg, HID, p.gate_w2, p.gate_b2, nullptr, 0, G2, HID, N_NODES, HID, HID, 2, stream);

    gate_combine<<<(nodeElems + 255) / 256, 256, 0, stream>>>(
        G2, CONV, skipp, ldsk, HB, nodeElems);

    // FFN with pre-LN and residual, written straight into X[:, (i+1)*64 : ]
    ln_rows<<<rowWaveBlocks, 256, 0, stream>>>(HB, HID, Gg, HID, p.n2_g, p.n2_b,
                                               N_NODES, HID, 0);
    gemm(Gg, HID, p.ffn_w1, p.ffn_b1, nullptr, 0, TMP256, 256, N_NODES, HID, 256, 3, stream);
    gemm(TMP256, 256, p.ffn_w2, p.ffn_b2, HB, HID, X + HID * (i + 1), XLD,
         N_NODES, 256, HID, 0, stream);
  }

  // output = xs[-1] = X[:, 256:320]
  copy_cols<<<(nodeElems + 255) / 256, 256, 0, stream>>>(
      X, XLD, HID * N_LAYERS, (float*)d_out, nodeElems);
}